// HBVMulTDET_69355131895810
// MI455X (gfx1250) — compile-verified
//
#include <hip/hip_runtime.h>
#include <hip/hip_bf16.h>
#include <stdint.h>

// ---------------- problem constants ----------------
#define NSTEP   2000
#define NGRID   2000
#define NMUL    4
#define NEARZERO 1e-5f

// ---------------- tiling ----------------
#define GB      64                 // grids per block
#define NTHREADS (GB * NMUL)       // 256 threads = 8 wave32
#define CHUNK   25                 // time steps staged per TDM transfer (2000/25 = 80 chunks)
#define NCHUNK  (NSTEP / CHUNK)
#define XSZ     (CHUNK * GB * 3)   // floats per x staging buffer  (4800 -> 19.2 KB)
#define QSZ     (CHUNK * GB)       // floats per q staging buffer  (1600 ->  6.4 KB)

typedef __attribute__((ext_vector_type(4))) unsigned int v4u;
typedef __attribute__((ext_vector_type(8))) int          v8i;
typedef __attribute__((ext_vector_type(4))) int          v4i;

// ---------------- TDM descriptor (CDNA5 ISA) ----------------
// 2D tile: tile_dim0 (contiguous floats) x tile_dim1 rows, row stride = stride0 elems.
struct TdmDesc { v4u g0; v8i g1; v4i g2; v4i g3; };

__device__ __forceinline__ TdmDesc tdm_make_desc(uint32_t lds_byte, uint64_t gaddr,
                                                 uint32_t tile0, uint32_t tile1,
                                                 uint32_t tdim0, uint32_t tdim1,
                                                 uint32_t stride0) {
  TdmDesc d;
  // group0: count=1 (valid user descriptor), lds_addr, 57-bit global addr, type=2
  d.g0.x = 1u;
  d.g0.y = lds_byte;
  d.g0.z = (uint32_t)gaddr;
  d.g0.w = ((uint32_t)(gaddr >> 32) & 0x01FFFFFFu) | (2u << 30);
  // group1: data_size=2 (4B), no multicast / barrier / iterate / pad
  d.g1[0] = (int)(2u << 16);
  d.g1[1] = (int)((tdim0 & 0xFFFFu) << 16);                               // tensor_dim0[15:0]
  d.g1[2] = (int)(((tdim0 >> 16) & 0xFFFFu) | ((tdim1 & 0xFFFFu) << 16)); // dim0 hi | dim1 lo
  d.g1[3] = (int)(((tdim1 >> 16) & 0xFFFFu) | ((tile0 & 0xFFFFu) << 16)); // dim1 hi | tile_dim0
  d.g1[4] = (int)(tile1 & 0xFFFFu);                                       // tile_dim1 (tile_dim2=0)
  d.g1[5] = (int)stride0;                                                 // tensor_dim0_stride lo32
  d.g1[6] = 0;                                                            // stride0 hi16 | stride1 lo16
  d.g1[7] = 0;                                                            // stride1 hi32
  d.g2 = 0;  // 2D tensor: groups 2/3 unused (tile/tensor dims 2..4 = 0)
  d.g3 = 0;
  return d;
}

__device__ __forceinline__ void tdm_load(uint32_t lds_byte, uint64_t gaddr,
                                         uint32_t tile0, uint32_t tile1,
                                         uint32_t tdim0, uint32_t tdim1, uint32_t stride0) {
  TdmDesc d = tdm_make_desc(lds_byte, gaddr, tile0, tile1, tdim0, tdim1, stride0);
#if __has_include(<hip/amd_detail/amd_gfx1250_TDM.h>)
  v8i z8 = 0;
  __builtin_amdgcn_tensor_load_to_lds(d.g0, d.g1, d.g2, d.g3, z8, 0);
#else
  __builtin_amdgcn_tensor_load_to_lds(d.g0, d.g1, d.g2, d.g3, 0);
#endif
}

__device__ __forceinline__ void tdm_store(uint32_t lds_byte, uint64_t gaddr,
                                          uint32_t tile0, uint32_t tile1,
                                          uint32_t tdim0, uint32_t tdim1, uint32_t stride0) {
  TdmDesc d = tdm_make_desc(lds_byte, gaddr, tile0, tile1, tdim0, tdim1, stride0);
#if __has_include(<hip/amd_detail/amd_gfx1250_TDM.h>)
  v8i z8 = 0;
  __builtin_amdgcn_tensor_store_from_lds(d.g0, d.g1, d.g2, d.g3, z8, 0);
#else
  __builtin_amdgcn_tensor_store_from_lds(d.g0, d.g1, d.g2, d.g3, 0);
#endif
}

// ---------------- HBV kernel ----------------
__global__ __launch_bounds__(NTHREADS)
void hbv_tdm_kernel(const float* __restrict__ x,          // [NSTEP, NGRID, 3]
                    const float* __restrict__ params_raw, // [NSTEP, NGRID, 14, NMUL]
                    float* __restrict__ out) {            // [NSTEP, NGRID]
  __shared__ float lds[2 * XSZ + 2 * QSZ];   // 51.2 KB

  const int tid    = threadIdx.x;
  const int blk    = blockIdx.x;
  const int gstart = blk * GB;
  const int gcount = (NGRID - gstart < GB) ? (NGRID - gstart) : GB;
  const int lg     = tid >> 2;        // local grid 0..63
  const int m      = tid & 3;         // nmul lane 0..3
  const int g      = gstart + lg;
  const bool active = (lg < gcount);
  const bool isWave0 = (tid < 32);    // wave-uniform predicate: only wave0 drives the TDM

  // ---- parameter scaling (last timestep only) ----
  const float lbv[14] = {1.0f, 50.f, 0.05f, 0.01f, 0.001f, 0.2f, 0.0f, 0.0f,
                         -2.5f, 0.5f, 0.0f, 0.0f, 0.3f, 0.0f};
  const float ubv[14] = {6.0f, 1000.f, 0.9f, 0.5f, 0.2f, 1.0f, 10.0f, 100.0f,
                         2.5f, 10.0f, 0.1f, 0.2f, 5.0f, 1.0f};
  float pp[14];
  if (active) {
    const float* pbase = params_raw
        + (size_t)(NSTEP - 1) * NGRID * 14 * NMUL
        + (size_t)g * 14 * NMUL + m;
#pragma unroll
    for (int i = 0; i < 14; ++i)
      pp[i] = pbase[i * NMUL] * (ubv[i] - lbv[i]) + lbv[i];
  } else {
#pragma unroll
    for (int i = 0; i < 14; ++i)       // benign mid-range values for idle lanes
      pp[i] = 0.5f * (lbv[i] + ubv[i]);
  }
  const float BETA = pp[0], FC = pp[1], K0 = pp[2], K1 = pp[3], K2 = pp[4];
  const float LP = pp[5], PERCp = pp[6], UZL = pp[7], TT = pp[8], CFMAX = pp[9];
  const float CFR = pp[10], CWH = pp[11], BETAET = pp[12], Cc = pp[13];
  const float invFC    = 1.0f / FC;
  const float invLPFC  = 1.0f / (LP * FC);
  const float CFRCFMAX = CFR * CFMAX;

  float SNOW = 0.001f, MELT = 0.001f, SM = 0.001f, SUZ = 0.001f, SLZ = 0.001f;

  const int      xrow = gcount * 3;                 // contiguous floats per step per block
  const uint64_t xg   = (uint64_t)(uintptr_t)x;
  const uint64_t og   = (uint64_t)(uintptr_t)out;

  // Prologue: async-stage chunk 0 forcing into LDS buffer 0 (TENSORcnt tracked).
  if (isWave0) {
    tdm_load(/*lds_byte=*/0u,
             xg + 4ull * ((uint64_t)gstart * 3),
             (uint32_t)xrow, CHUNK, NGRID * 3, NSTEP, NGRID * 3);
  }

  for (int c = 0; c < NCHUNK; ++c) {
    const int cur = c & 1, nxt = cur ^ 1;

    // Wait for chunk c's load. After c==0, allow the latest tensor_store to
    // remain in flight (tensor ops complete in order, so <=1 implies the
    // pending load and all older stores are done).
    if (isWave0) {
      if (c == 0) __builtin_amdgcn_s_wait_tensorcnt(0);
      else        __builtin_amdgcn_s_wait_tensorcnt(1);
    }
    __syncthreads();   // chunk-c forcing now visible block-wide

    // Software pipeline: kick off chunk c+1 load into the other buffer.
    if (isWave0 && (c + 1 < NCHUNK)) {
      const uint64_t t0n = (uint64_t)(c + 1) * CHUNK;
      tdm_load((uint32_t)(nxt * XSZ * 4),
               xg + 4ull * (t0n * (NGRID * 3) + (uint64_t)gstart * 3),
               (uint32_t)xrow, CHUNK, NGRID * 3, NSTEP, NGRID * 3);
    }

    const float* xbuf = lds + cur * XSZ;
    float*       qbuf = lds + 2 * XSZ + cur * QSZ;

#pragma unroll 1
    for (int r = 0; r < CHUNK; ++r) {
      const float* row = xbuf + r * xrow + lg * 3;
      const float P = row[0], T = row[1], PET = row[2];

      // ---- HBV step (matches reference scan body) ----
      const float rain = (T >= TT) ? P : 0.0f;
      const float snow = P - rain;
      SNOW += snow;
      float melt = fminf(fmaxf(CFMAX * (T - TT), 0.0f), SNOW);
      MELT += melt;  SNOW -= melt;
      float refreeze = fminf(fmaxf(CFRCFMAX * (TT - T), 0.0f), MELT);
      SNOW += refreeze;  MELT -= refreeze;
      float tosoil = fmaxf(MELT - CWH * SNOW, 0.0f);
      MELT -= tosoil;
      float sw = fminf(fmaxf(__powf(SM * invFC, BETA), 0.0f), 1.0f);
      float recharge = (rain + tosoil) * sw;
      SM += rain + tosoil - recharge;
      float excess = fmaxf(SM - FC, 0.0f);
      SM -= excess;
      float ef = fminf(fmaxf(__powf(SM * invLPFC, BETAET), 0.0f), 1.0f);
      float ETact = fminf(SM, PET * ef);
      SM = fmaxf(SM - ETact, NEARZERO);
      float capillary = fminf(SLZ, Cc * SLZ * (1.0f - fminf(SM * invFC, 1.0f)));
      SM  = fmaxf(SM + capillary, NEARZERO);
      SLZ = fmaxf(SLZ - capillary, NEARZERO);
      SUZ += recharge + excess;
      float PERCa = fminf(SUZ, PERCp);
      SUZ -= PERCa;
      float Q0 = K0 * fmaxf(SUZ - UZL, 0.0f);
      SUZ -= Q0;
      float Q1 = K1 * SUZ;
      SUZ -= Q1;
      SLZ += PERCa;
      float Q2 = K2 * SLZ;
      SLZ -= Q2;
      float q = Q0 + Q1 + Q2;

      // mean over NMUL: sum across the 4-lane group (wave32 shuffle), lane m==0 stages
      q += __shfl_xor(q, 1, 32);
      q += __shfl_xor(q, 2, 32);
      if (active && m == 0)
        qbuf[r * gcount + lg] = q * 0.25f;
    }

    __syncthreads();   // staging complete before TDM reads it

    // Async-drain chunk c's Q tile: LDS -> out[t0..t0+CHUNK, gstart..gstart+gcount)
    if (isWave0) {
      const uint64_t t0 = (uint64_t)c * CHUNK;
      tdm_store((uint32_t)((2 * XSZ + cur * QSZ) * 4),
                og + 4ull * (t0 * NGRID + (uint64_t)gstart),
                (uint32_t)gcount, CHUNK, NGRID, NSTEP, NGRID);
    }
  }

  if (isWave0) __builtin_amdgcn_s_wait_tensorcnt(0);  // (S_ENDPGM also implies wait-idle)
}

// ---------------- host entry ----------------
extern "C" void kernel_launch(void* const* d_in, const int* in_sizes, int n_in,
                              void* d_out, int out_size, void* d_ws, size_t ws_size,
                              hipStream_t stream) {
  (void)in_sizes; (void)n_in; (void)out_size; (void)d_ws; (void)ws_size;
  const float* x          = (const float*)d_in[0];  // x_hydro_model  [NSTEP,NGRID,3]
  // d_in[1] = c_hydro_model [NGRID,2] — unused (pet_module='dataset')
  const float* params_raw = (const float*)d_in[2];  // [NSTEP,NGRID,14,NMUL]
  float*       out        = (float*)d_out;          // [NSTEP,NGRID]

  dim3 grid((NGRID + GB - 1) / GB);   // 32 blocks
  dim3 block(NTHREADS);               // 256 threads = 8 wave32
  hipLaunchKernelGGL(hbv_tdm_kernel, grid, block, 0, stream, x, params_raw, out);
}